// pRNN_25950192402502
// MI455X (gfx1250) — compile-verified
//
#include <hip/hip_runtime.h>
#include <hip/hip_bf16.h>

// Problem: B=8192, I=256, N=64, H=64, D=128, SRC=4352.
// Only out[:, -1] (n = 63) is returned, so we compute just:
//   g[b,d]   = flat[b, idx[63,d]]        (flat = [relu(x*w_in+b_in) | h_state^T])
//   out[b,o] = relu( sum_d g[b,d]*W[63,o,d] + bias[63,o] )
// FP32 WMMA (V_WMMA_F32_16X16X4_F32) keeps exact f32 semantics; the kernel is
// HBM-bound (~6-30 MB of traffic), so low-precision WMMA would buy nothing.

typedef float v2f __attribute__((ext_vector_type(2)));
typedef float v8f __attribute__((ext_vector_type(8)));

#define BDIM 8192
#define IDIM 256
#define HDIM 64
#define DDIM 128
#define ROWS 32            // output rows (b) per workgroup
#define SG_STRIDE 40       // sG[d][r], r in [0,32); +2-row half-wave offset -> disjoint banks
#define SW_STRIDE 72       // sW[d][o], o in [0,64); 2*72 mod 64 = 16 -> disjoint banks

__global__ __launch_bounds__(256) void prnn_last_head_kernel(
    const float* __restrict__ x,        // (B, I)
    const float* __restrict__ h_state,  // (N, B, H)
    const float* __restrict__ w_in,     // (I)
    const float* __restrict__ b_in,     // (I)
    const float* __restrict__ W,        // (N, H, D)
    const float* __restrict__ bias,     // (N, H)
    const int*   __restrict__ idx,      // (N, D)
    float*       __restrict__ out)      // (B, H)
{
    __shared__ float sG[DDIM * SG_STRIDE]; // gathered tile, transposed: [d][row]
    __shared__ float sW[DDIM * SW_STRIDE]; // W[63] transposed:          [d][o]

    const float* WN   = W    + 63 * HDIM * DDIM;
    const float* bN   = bias + 63 * HDIM;
    const int*   idxN = idx  + 63 * DDIM;

    const int tid  = threadIdx.x;
    const int row0 = blockIdx.x * ROWS;

    // ---- Stage W[63] into LDS (coalesced global reads: i walks d fastest) ----
    for (int i = tid; i < HDIM * DDIM; i += 256) {
        const int o = i >> 7;        // [0,64)
        const int d = i & (DDIM - 1);
        sW[d * SW_STRIDE + o] = WN[i]; // WN[o*128 + d]
    }

    // ---- Gather 32 rows x 128 cols into LDS (one d per 32-lane wave => uniform branch) ----
    for (int i = tid; i < ROWS * DDIM; i += 256) {
        const int d  = i >> 5;       // [0,128)
        const int r  = i & (ROWS - 1);
        const int bb = row0 + r;
        const int s  = idxN[d];
        float v;
        if (s < IDIM) {
            v = fmaxf(fmaf(x[bb * IDIM + s], w_in[s], b_in[s]), 0.0f);
        } else {
            const int q = s - IDIM;            // q = n'*64 + h'
            v = h_state[(q >> 6) * (BDIM * HDIM) + bb * HDIM + (q & 63)];
        }
        sG[d * SG_STRIDE + r] = v;             // lanes write 32 consecutive banks
    }

    __syncthreads();

    // ---- 8 waves: 2 row-tiles x 4 col-tiles of 16x16 ----
    const int wave = tid >> 5;
    const int lane = tid & 31;
    const int half = lane >> 4;      // 0: K=k,k+1   1: K=k+2,k+3
    const int l    = lane & 15;
    const int m0   = (wave & 1) * 16;
    const int n0   = (wave >> 1) * 16;

    v8f acc = {};
    for (int k0 = 0; k0 < DDIM; k0 += 4) {
        const int ka = k0 + 2 * half;
        v2f a, bf;
        // A 16x4 f32 layout: lane l holds (M=m0+l, K=ka) in v0, (M=m0+l, K=ka+1) in v1
        a.x  = sG[ ka      * SG_STRIDE + m0 + l];
        a.y  = sG[(ka + 1) * SG_STRIDE + m0 + l];
        // B 4x16 f32 layout (symmetric, N on lanes): lane l holds (K=ka, N=n0+l), (K=ka+1, N=n0+l)
        bf.x = sW[ ka      * SW_STRIDE + n0 + l];
        bf.y = sW[(ka + 1) * SW_STRIDE + n0 + l];
        acc = __builtin_amdgcn_wmma_f32_16x16x4_f32(
            /*neg_a=*/false, a, /*neg_b=*/false, bf,
            /*c_mod=*/(short)0, acc, /*reuse_a=*/false, /*reuse_b=*/false);
    }

    // ---- Epilogue: +bias[63], relu (n=63 >= 1 so relu applies), store ----
    const float bv = bN[n0 + l];
    #pragma unroll
    for (int r = 0; r < 8; ++r) {
        // C/D 16x16 f32 layout: VGPR r = row m0 + 8*half + r, col n0 + l
        const int row = row0 + m0 + half * 8 + r;
        out[row * HDIM + n0 + l] = fmaxf(acc[r] + bv, 0.0f);
    }
}

extern "C" void kernel_launch(void* const* d_in, const int* in_sizes, int n_in,
                              void* d_out, int out_size, void* d_ws, size_t ws_size,
                              hipStream_t stream) {
    const float* x       = (const float*)d_in[0];
    const float* h_state = (const float*)d_in[1];
    const float* w_in    = (const float*)d_in[2];
    const float* b_in    = (const float*)d_in[3];
    const float* W       = (const float*)d_in[4];
    const float* bias    = (const float*)d_in[5];
    const int*   idx     = (const int*)d_in[6];
    float*       out     = (float*)d_out;

    dim3 grid(BDIM / ROWS);   // 256 workgroups
    dim3 block(256);          // 8 wave32s
    prnn_last_head_kernel<<<grid, block, 0, stream>>>(
        x, h_state, w_in, b_in, W, bias, idx, out);
}